// RelativeAttention_1580547974434
// MI455X (gfx1250) — compile-verified
//
#include <hip/hip_runtime.h>
#include <math.h>

#define HEADS  12
#define SEQ    2048
#define DEPTH  64
#define NSTATE 768
#define BATCH  2

typedef float f32x2 __attribute__((ext_vector_type(2)));
typedef float f32x8 __attribute__((ext_vector_type(8)));

// D = A(16x4 f32) * B(4x16 f32) + C(16x16 f32)  -> v_wmma_f32_16x16x4_f32
__device__ __forceinline__ f32x8 wmma_f32(f32x2 a, f32x2 b, f32x8 c) {
  return __builtin_amdgcn_wmma_f32_16x16x4_f32(false, a, false, b, (short)0, c,
                                               false, false);
}

// Async wave-wide 128b copy global -> LDS (ASYNCcnt path, CDNA5).
// loff: per-lane LDS byte address, goff: per-lane byte offset from base.
__device__ __forceinline__ void async_b128(unsigned loff, unsigned goff,
                                           const float* base) {
  asm volatile("global_load_async_to_lds_b128 %0, %1, %2"
               :: "v"(loff), "v"(goff), "s"(base) : "memory");
}
__device__ __forceinline__ void wait_async0() {
  asm volatile("s_wait_asynccnt 0x0" ::: "memory");
}
// Low 32 bits of a generic pointer into the LDS aperture == LDS byte offset.
__device__ __forceinline__ unsigned lds_of(const void* p) {
  return (unsigned)(unsigned long long)(unsigned long long)(uintptr_t)p;
}

// ---------------------------------------------------------------------------
// Kernel 1: c = x @ w_attn + b_attn, scattered directly into q/k/v [B,H,S,D]
// One wave computes a 16x64 output tile (4 WMMA accumulators).
// ---------------------------------------------------------------------------
__global__ void qkv_gemm(const float* __restrict__ x, const float* __restrict__ w,
                         const float* __restrict__ bias,
                         float* __restrict__ q, float* __restrict__ k,
                         float* __restrict__ v) {
  const int rt = blockIdx.x & 255;         // 4096/16 row tiles
  const int cg = blockIdx.x >> 8;          // 2304/64 col groups
  const int i0 = rt * 16, j0 = cg * 64;
  const int lane = threadIdx.x;
  const int mr = lane & 15;                // A row / B col within tile
  const int kb = (lane >> 4) * 2;          // K sub-lane offset
  const int half = lane >> 4;

  f32x8 acc[4] = {};
  for (int kk = 0; kk < NSTATE; kk += 4) {
    f32x2 af = *(const f32x2*)(x + (size_t)(i0 + mr) * NSTATE + kk + kb);
#pragma unroll
    for (int f = 0; f < 4; ++f) {
      const int n = j0 + f * 16 + mr;
      f32x2 bf;
      bf.x = w[(size_t)(kk + kb) * (3 * NSTATE) + n];
      bf.y = w[(size_t)(kk + kb + 1) * (3 * NSTATE) + n];
      acc[f] = wmma_f32(af, bf, acc[f]);
    }
  }

  const int which = j0 / NSTATE;                 // 0=q 1=k 2=v (uniform)
  const int hidx  = (j0 % NSTATE) >> 6;          // head (uniform)
  float* dst = (which == 0) ? q : ((which == 1) ? k : v);
#pragma unroll
  for (int f = 0; f < 4; ++f)
#pragma unroll
    for (int r = 0; r < 8; ++r) {
      const int M   = r + half * 8;
      const int row = i0 + M;
      const int b   = row >> 11, s = row & 2047;
      const int col = j0 + f * 16 + mr;
      dst[(((size_t)b * HEADS + hidx) * SEQ + s) * DEPTH + (col & 63)] =
          acc[f][r] + bias[col];
    }
}

// ---------------------------------------------------------------------------
// Kernel 2: fused attention with skewed relative bias, flash-style.
// 4 waves per workgroup, each owning a 16-query-row block of the SAME (b,h);
// they share a double-buffered k/v tile staged via async-to-LDS DMA.
// Skew semantics (matches pad/reshape/slice exactly):
//   t <= s   : q[s]  . E[S-1-s+t]
//   t == s+1 : 0
//   t >= s+2 : q[s+1]. E[t-s-2]    (wrap-around values; reference has no mask)
// LDS row pitch = 68 floats to spread B-fragment ds_load_b64 across banks.
// ---------------------------------------------------------------------------
#define KV_PITCH 68
#define KV_TILE  (16 * KV_PITCH)                 // 1088 floats
__global__ void attn_kernel(const float* __restrict__ q, const float* __restrict__ k,
                            const float* __restrict__ v, const float* __restrict__ E,
                            float* __restrict__ out) {
  __shared__ __align__(16) float smem[4 * KV_TILE + 4 * 256];
  // [0]        k buf0   [KV_TILE]  k buf1
  // [2*KV_TILE] v buf0  [3*KV_TILE] v buf1
  // [4*KV_TILE + wv*256] per-wave p tile

  const int wv   = threadIdx.x >> 5;
  const int lane = threadIdx.x & 31;
  const int sb4  = blockIdx.x & 31;            // 32 groups of 4 query blocks
  const int bh   = blockIdx.x >> 5;            // b*HEADS + h
  const int h    = bh % HEADS;
  const int s0   = (sb4 * 4 + wv) * 16;
  const float* qb  = q + (size_t)bh * SEQ * DEPTH;
  const float* kb_ = k + (size_t)bh * SEQ * DEPTH;
  const float* vb  = v + (size_t)bh * SEQ * DEPTH;
  const float* Eh  = E + (size_t)h * SEQ * DEPTH;   // MAX_SEQ == SEQ

  const int mr   = lane & 15;
  const int kb2  = (lane >> 4) * 2;
  const int half = lane >> 4;

  // DMA lane addressing: instruction i moves rows {2i, 2i+1} of a 16x64 tile.
  const unsigned rip = lane >> 4, seg = lane & 15;
  const unsigned goff0 = rip * 256u + seg * 16u;                  // global bytes
  const unsigned loff0 = rip * (KV_PITCH * 4u) + seg * 16u;       // LDS bytes
  const unsigned klds[2] = { lds_of(&smem[0]), lds_of(&smem[KV_TILE]) };
  const unsigned vlds[2] = { lds_of(&smem[2 * KV_TILE]), lds_of(&smem[3 * KV_TILE]) };

  // each wave issues its 4 of the 16 async b128 per tile pair
  auto stage = [&](int t0, int buf) {
    const float* kg = kb_ + (size_t)t0 * DEPTH;
    const float* vg = vb + (size_t)t0 * DEPTH;
#pragma unroll
    for (int ii = 0; ii < 2; ++ii) {
      const unsigned i = (unsigned)(wv + ii * 4);
      async_b128(klds[buf] + i * (2 * KV_PITCH * 4u) + loff0,
                 i * 512u + goff0, kg);
      async_b128(vlds[buf] + i * (2 * KV_PITCH * 4u) + loff0,
                 i * 512u + goff0, vg);
    }
  };

  // q tile as 16 A-fragments (K=64), plus row-shifted copy for t>=s+2 region
  f32x2 qa[16], qa2[16];
#pragma unroll
  for (int c = 0; c < 16; ++c)
    qa[c] = *(const f32x2*)(qb + (size_t)(s0 + mr) * DEPTH + c * 4 + kb2);
  int srow2 = s0 + 1 + mr; if (srow2 > SEQ - 1) srow2 = SEQ - 1;
#pragma unroll
  for (int c = 0; c < 16; ++c)
    qa2[c] = *(const f32x2*)(qb + (size_t)srow2 * DEPTH + c * 4 + kb2);

  f32x8 oacc[4] = {};
  float mrun[8], lrun[8];
#pragma unroll
  for (int r = 0; r < 8; ++r) { mrun[r] = -3.0e38f; lrun[r] = 0.0f; }

  stage(0, 0);
  wait_async0();
  __syncthreads();

  for (int t0 = 0; t0 < SEQ; t0 += 16) {
    const int buf = (t0 >> 4) & 1;
    const bool hasNext = (t0 + 16) < SEQ;
    if (hasNext) stage(t0 + 16, buf ^ 1);      // overlap DMA with compute

    const float* kt = smem + buf * KV_TILE;
    const float* vt = smem + 2 * KV_TILE + buf * KV_TILE;
    float* pb = smem + 4 * KV_TILE + wv * 256;

    // ---- scores = q . k^T (16 chained f32 WMMAs over K=64), k from LDS
    f32x8 sc = {};
#pragma unroll
    for (int c = 0; c < 16; ++c) {
      f32x2 bf = *(const f32x2*)(kt + mr * KV_PITCH + c * 4 + kb2);
      sc = wmma_f32(qa[c], bf, sc);
    }

    // ---- relative bias (region flags are wave-uniform)
    const bool anyBelow = (t0 <= s0 + 15);
    const bool anyAbove = (t0 + 15 >= s0 + 2);
    f32x8 rel = {};
    if (anyBelow) {
      const int j0 = SEQ - 16 - s0 + t0;       // E window base, jj in [0,30]
      int e0 = j0 + mr;       e0 = e0 < 0 ? 0 : (e0 > SEQ - 1 ? SEQ - 1 : e0);
      int e1 = j0 + 16 + mr;  e1 = e1 < 0 ? 0 : (e1 > SEQ - 1 ? SEQ - 1 : e1);
      f32x8 g0 = {}, g1 = {};
#pragma unroll
      for (int c = 0; c < 16; ++c) {
        f32x2 b0 = *(const f32x2*)(Eh + (size_t)e0 * DEPTH + c * 4 + kb2);
        f32x2 b1 = *(const f32x2*)(Eh + (size_t)e1 * DEPTH + c * 4 + kb2);
        g0 = wmma_f32(qa[c], b0, g0);
        g1 = wmma_f32(qa[c], b1, g1);
      }
#pragma unroll
      for (int r = 0; r < 8; ++r) {
        const int M = r + half * 8;
        const int jj = 15 - M + mr;                       // [0,30]
        const int srcLane = (half << 4) | (jj & 15);
        const float v0 = __shfl(g0[r], srcLane, 32);
        const float v1 = __shfl(g1[r], srcLane, 32);
        const float gv = (jj < 16) ? v0 : v1;
        rel[r] = ((t0 + mr) <= (s0 + M)) ? gv : rel[r];
      }
    }
    if (anyAbove) {
      const int j20 = t0 - s0 - 17;
      int e0 = j20 + mr;       e0 = e0 < 0 ? 0 : (e0 > SEQ - 1 ? SEQ - 1 : e0);
      int e1 = j20 + 16 + mr;  e1 = e1 < 0 ? 0 : (e1 > SEQ - 1 ? SEQ - 1 : e1);
      f32x8 g0 = {}, g1 = {};
#pragma unroll
      for (int c = 0; c < 16; ++c) {
        f32x2 b0 = *(const f32x2*)(Eh + (size_t)e0 * DEPTH + c * 4 + kb2);
        f32x2 b1 = *(const f32x2*)(Eh + (size_t)e1 * DEPTH + c * 4 + kb2);
        g0 = wmma_f32(qa2[c], b0, g0);
        g1 = wmma_f32(qa2[c], b1, g1);
      }
#pragma unroll
      for (int r = 0; r < 8; ++r) {
        const int M = r + half * 8;
        const int jj = 15 - M + mr;
        const int srcLane = (half << 4) | (jj & 15);
        const float v0 = __shfl(g0[r], srcLane, 32);
        const float v1 = __shfl(g1[r], srcLane, 32);
        const float gv = (jj < 16) ? v0 : v1;
        rel[r] = ((t0 + mr) >= (s0 + M + 2)) ? gv : rel[r];
      }
    }
#pragma unroll
    for (int r = 0; r < 8; ++r) sc[r] = (sc[r] + rel[r]) * 0.125f; // 1/sqrt(64)

    // ---- online softmax: row lives in one VGPR across a 16-lane group
    float alpha[8];
#pragma unroll
    for (int r = 0; r < 8; ++r) {
      float mx = sc[r];
      mx = fmaxf(mx, __shfl_xor(mx, 1, 16));
      mx = fmaxf(mx, __shfl_xor(mx, 2, 16));
      mx = fmaxf(mx, __shfl_xor(mx, 4, 16));
      mx = fmaxf(mx, __shfl_xor(mx, 8, 16));
      const float mn = fmaxf(mrun[r], mx);
      alpha[r] = __expf(mrun[r] - mn);
      mrun[r] = mn;
      const float p = __expf(sc[r] - mn);
      sc[r] = p;
      float sm = p;
      sm += __shfl_xor(sm, 1, 16);
      sm += __shfl_xor(sm, 2, 16);
      sm += __shfl_xor(sm, 4, 16);
      sm += __shfl_xor(sm, 8, 16);
      lrun[r] = lrun[r] * alpha[r] + sm;
    }
#pragma unroll
    for (int f = 0; f < 4; ++f)
#pragma unroll
      for (int r = 0; r < 8; ++r) oacc[f][r] = oacc[f][r] * alpha[r];

    // ---- relayout p (C-frag -> A-frag) through per-wave LDS scratch
    __syncthreads();
#pragma unroll
    for (int r = 0; r < 8; ++r) pb[(r + half * 8) * 16 + mr] = sc[r];
    __syncthreads();
    f32x2 pa[4];
#pragma unroll
    for (int c = 0; c < 4; ++c) {
      pa[c].x = pb[mr * 16 + c * 4 + kb2];
      pa[c].y = pb[mr * 16 + c * 4 + kb2 + 1];
    }

    // ---- oacc += p(16x16) @ v(16x64), v from LDS
#pragma unroll
    for (int f = 0; f < 4; ++f)
#pragma unroll
      for (int c = 0; c < 4; ++c) {
        f32x2 bf;
        bf.x = vt[(c * 4 + kb2) * KV_PITCH + f * 16 + mr];
        bf.y = vt[(c * 4 + kb2 + 1) * KV_PITCH + f * 16 + mr];
        oacc[f] = wmma_f32(pa[c], bf, oacc[f]);
      }

    if (hasNext) {           // own DMA done + publish to sibling waves
      wait_async0();
      __syncthreads();
    }
  }

#pragma unroll
  for (int r = 0; r < 8; ++r) lrun[r] = 1.0f / lrun[r];
#pragma unroll
  for (int f = 0; f < 4; ++f)
#pragma unroll
    for (int r = 0; r < 8; ++r) {
      const int M = r + half * 8;
      out[((size_t)bh * SEQ + s0 + M) * DEPTH + f * 16 + mr] =
          oacc[f][r] * lrun[r];
    }
}

// ---------------------------------------------------------------------------
// Kernel 3: out = a @ w_proj + b_proj ; a is [B,H,S,D] == logical [B*S, 768]
// ---------------------------------------------------------------------------
__global__ void proj_gemm(const float* __restrict__ a, const float* __restrict__ w,
                          const float* __restrict__ bias, float* __restrict__ out) {
  const int rt = blockIdx.x & 255;
  const int cg = blockIdx.x >> 8;          // 768/64 col groups
  const int i0 = rt * 16, j0 = cg * 64;
  const int lane = threadIdx.x;
  const int mr = lane & 15, kb = (lane >> 4) * 2, half = lane >> 4;
  const int row = i0 + mr;
  const int b = row >> 11, s = row & 2047;

  f32x8 acc[4] = {};
  for (int kk = 0; kk < NSTATE; kk += 4) {
    const int hidx = kk >> 6, d = (kk & 63) + kb;
    f32x2 af = *(const f32x2*)(a + (((size_t)b * HEADS + hidx) * SEQ + s) * DEPTH + d);
#pragma unroll
    for (int f = 0; f < 4; ++f) {
      const int n = j0 + f * 16 + mr;
      f32x2 bf;
      bf.x = w[(size_t)(kk + kb) * NSTATE + n];
      bf.y = w[(size_t)(kk + kb + 1) * NSTATE + n];
      acc[f] = wmma_f32(af, bf, acc[f]);
    }
  }
#pragma unroll
  for (int f = 0; f < 4; ++f)
#pragma unroll
    for (int r = 0; r < 8; ++r) {
      const int M = r + half * 8;
      const int col = j0 + f * 16 + mr;
      out[(size_t)(i0 + M) * NSTATE + col] = acc[f][r] + bias[col];
    }
}

// ---------------------------------------------------------------------------
extern "C" void kernel_launch(void* const* d_in, const int* in_sizes, int n_in,
                              void* d_out, int out_size, void* d_ws, size_t ws_size,
                              hipStream_t stream) {
  const float* x      = (const float*)d_in[0];
  const float* w_attn = (const float*)d_in[1];
  const float* b_attn = (const float*)d_in[2];
  const float* w_proj = (const float*)d_in[3];
  const float* b_proj = (const float*)d_in[4];
  const float* E      = (const float*)d_in[5];
  float* out = (float*)d_out;

  const size_t per = (size_t)BATCH * HEADS * SEQ * DEPTH;   // 3,145,728 floats
  float* q = (float*)d_ws;
  float* k = q + per;
  float* v = k + per;
  float* a = v + per;

  qkv_gemm<<<dim3(256 * 36), 32, 0, stream>>>(x, w_attn, b_attn, q, k, v);
  attn_kernel<<<dim3(BATCH * HEADS * (SEQ / 64)), 128, 0, stream>>>(q, k, v, E, a);
  proj_gemm<<<dim3(256 * 12), 32, 0, stream>>>(a, w_proj, b_proj, out);
}